// TorchSumProdLayer_78262894068504
// MI455X (gfx1250) — compile-verified
//
#include <hip/hip_runtime.h>
#include <hip/hip_bf16.h>

typedef __attribute__((ext_vector_type(2))) float v2f;
typedef __attribute__((ext_vector_type(8))) float v8f;

#if defined(__has_builtin)
#if __has_builtin(__builtin_amdgcn_wmma_f32_16x16x4_f32)
#define SPN_HAVE_WMMA 1
#endif
#endif
#ifndef SPN_HAVE_WMMA
#define SPN_HAVE_WMMA 0
#endif

namespace {
constexpr int BATCH   = 256;
constexpr int D_IN    = 8192;
constexpr int N_SUM   = 4096;
constexpr int KCH     = 16;              // children per sum node
constexpr int R_ROWS  = N_SUM * KCH;     // 65536
constexpr int N_PASS  = 2048;
constexpr int N_NODES = N_SUM + N_PASS;  // 6144
}

// ---------------------------------------------------------------------------
// Kernel 0: per-node log-softmax of the child weights, done once.
//   logw[n*16+k] = w[n*16+k] - logsumexp_k(w[n*16+..])
// One thread per node; each thread touches a contiguous 64B chunk of w/logw.
// ---------------------------------------------------------------------------
__global__ void __launch_bounds__(256) spn_logw(const float* __restrict__ w,
                                                float* __restrict__ logw) {
  const int node = blockIdx.x * blockDim.x + threadIdx.x;
  if (node >= N_SUM) return;
  const float* wr = w + (size_t)node * KCH;
  float wv[KCH];
#pragma unroll
  for (int k = 0; k < KCH; ++k) wv[k] = wr[k];
  float m = wv[0];
#pragma unroll
  for (int k = 1; k < KCH; ++k) m = fmaxf(m, wv[k]);
  float s = 0.f;
#pragma unroll
  for (int k = 0; k < KCH; ++k) s += __expf(wv[k] - m);
  const float lse = m + __logf(s);
  float* lw = logw + (size_t)node * KCH;
#pragma unroll
  for (int k = 0; k < KCH; ++k) lw[k] = wv[k] - lse;
}

// ---------------------------------------------------------------------------
// Kernel 1: transpose x (BATCH x D_IN) -> xT (D_IN x BATCH) so per-column
// gathers in the main kernel are contiguous across the batch dimension.
// grid (D_IN/32, BATCH/32) = (256, 8), block 256 (32x8).
// ---------------------------------------------------------------------------
__global__ void __launch_bounds__(256) spn_transpose(const float* __restrict__ x,
                                                     float* __restrict__ xT) {
  __shared__ float t[32][33];
  const int d0 = blockIdx.x * 32;
  const int b0 = blockIdx.y * 32;
  const int tx = threadIdx.x & 31;
  const int ty = threadIdx.x >> 5;
#pragma unroll
  for (int i = 0; i < 32; i += 8)
    t[ty + i][tx] = x[(size_t)(b0 + ty + i) * D_IN + (d0 + tx)];
  __syncthreads();
#pragma unroll
  for (int i = 0; i < 32; i += 8)
    xT[(size_t)(d0 + ty + i) * BATCH + (b0 + tx)] = t[tx][ty + i];
}

// ---------------------------------------------------------------------------
// Kernel 2: fused sum-node forward.
//   One wave32 handles two sum nodes (one per pass) x 16 batches. Lane l owns batch l%16; the
//   lane pair (l, l+16) splits the 16 child rows as k = 4j + 2*(l>>4) + {0,1},
//   matching the K-striping of the V_WMMA_F32_16X16X4_F32 B operand, so the
//   segmented sum of exp() runs as 4 chained WMMAs with A = ones(16x4):
//   D[m][n] = sum_k E[k][n] = s[bl] replicated in every D VGPR of lane l.
//   Block = 8 waves -> 16 nodes x 16 batches; result tile staged in LDS and
//   written coalesced along the node axis.
// grid (N_SUM/16, BATCH/16) = (256, 16), block 256.
// ---------------------------------------------------------------------------
__global__ void __launch_bounds__(256) spn_sum_nodes(
    const float* __restrict__ xT, const float* __restrict__ x, int use_xt,
    const float* __restrict__ w, const float* __restrict__ logw, int have_logw,
    const int* __restrict__ edge_col, const float* __restrict__ edge_val,
    float* __restrict__ out) {
  __shared__ float tile[16 * 17];

  const int tid  = threadIdx.x;
  const int wave = tid >> 5;
  const int lane = tid & 31;
  const int bl   = lane & 15;   // batch within tile
  const int half = lane >> 4;   // which half of the K striping this lane feeds
  const int n0   = blockIdx.x * 16;
  const int b0   = blockIdx.y * 16;
  const int b    = b0 + bl;

#pragma unroll
  for (int nn = 0; nn < 2; ++nn) {
    const int nloc = wave * 2 + nn;
    const int node = n0 + nloc;

    float yr[8], lw[8];
#pragma unroll
    for (int j = 0; j < 4; ++j) {
#pragma unroll
      for (int i = 0; i < 2; ++i) {
        const int k  = 4 * j + 2 * half + i;
        const int r  = node * KCH + k;
        const int e0 = r * 2;
        const int c0 = edge_col[e0];
        const int c1 = edge_col[e0 + 1];
        const float v0 = edge_val[e0];
        const float v1 = edge_val[e0 + 1];
        float xv0, xv1;
        if (use_xt) {  // contiguous 64B segments across the 16 batch lanes
          xv0 = xT[(size_t)c0 * BATCH + b];
          xv1 = xT[(size_t)c1 * BATCH + b];
        } else {       // fallback: strided gather straight from x
          xv0 = x[(size_t)b * D_IN + c0];
          xv1 = x[(size_t)b * D_IN + c1];
        }
        yr[2 * j + i] = fmaf(xv0, v0, xv1 * v1);
        lw[2 * j + i] = have_logw ? logw[r] : w[r];
      }
    }

    if (!have_logw) {
      // inline per-node log-softmax normalizer over the 16 child weights
      float mW = lw[0];
#pragma unroll
      for (int t = 1; t < 8; ++t) mW = fmaxf(mW, lw[t]);
      mW = fmaxf(mW, __shfl_xor(mW, 16, 32));
      float sW = 0.f;
#pragma unroll
      for (int t = 0; t < 8; ++t) sW += __expf(lw[t] - mW);
      sW += __shfl_xor(sW, 16, 32);
      const float lseW = mW + __logf(sW);
#pragma unroll
      for (int t = 0; t < 8; ++t) lw[t] -= lseW;
    }

    float y[8];
#pragma unroll
    for (int t = 0; t < 8; ++t) y[t] = yr[t] + lw[t];

    // stable logsumexp over the node's 16 rows for this batch lane-pair
    float m = y[0];
#pragma unroll
    for (int t = 1; t < 8; ++t) m = fmaxf(m, y[t]);
    m = fmaxf(m, __shfl_xor(m, 16, 32));

    float e[8];
#pragma unroll
    for (int t = 0; t < 8; ++t) e[t] = __expf(y[t] - m);

#if SPN_HAVE_WMMA
    // D = ones(16x4) x E(4x16) + C, chained over 4 K-chunks:
    // every D VGPR of lane l ends up holding s[bl] = sum_k exp(y[k]-m).
    v2f aones = {1.0f, 1.0f};
    v8f acc = {};
#pragma unroll
    for (int j = 0; j < 4; ++j) {
      v2f bm = {e[2 * j], e[2 * j + 1]};
      acc = __builtin_amdgcn_wmma_f32_16x16x4_f32(
          false, aones, false, bm, (short)0, acc, false, false);
    }
    float s = acc[0];
#else
    float s = 0.f;
#pragma unroll
    for (int t = 0; t < 8; ++t) s += e[t];
    s += __shfl_xor(s, 16, 32);
#endif

    const float ll = m + __logf(s);
    if (half == 0) tile[nloc * 17 + bl] = ll;
  }

  __syncthreads();
  // coalesced write of the 16(b) x 16(node) tile: consecutive tid -> node axis
  const int nc = tid & 15;
  const int br = tid >> 4;
  out[(size_t)(b0 + br) * N_NODES + (n0 + nc)] = tile[nc * 17 + br];
}

// ---------------------------------------------------------------------------
// Kernel 3: pass-through nodes  out[b, scopes_out[j]] = x[b, scopes_in[j]]
// grid (N_PASS/256, BATCH), block 256. Writes coalesced; reads are 4B gathers
// that hit the L2-resident x.
// ---------------------------------------------------------------------------
__global__ void __launch_bounds__(256) spn_pass(const float* __restrict__ x,
                                                const int* __restrict__ scopes_in,
                                                const int* __restrict__ scopes_out,
                                                float* __restrict__ out) {
  const int j = blockIdx.x * blockDim.x + threadIdx.x;
  const int b = blockIdx.y;
  if (j < N_PASS)
    out[(size_t)b * N_NODES + scopes_out[j]] = x[(size_t)b * D_IN + scopes_in[j]];
}

extern "C" void kernel_launch(void* const* d_in, const int* in_sizes, int n_in,
                              void* d_out, int out_size, void* d_ws, size_t ws_size,
                              hipStream_t stream) {
  const float* x          = (const float*)d_in[0];  // (BATCH, D_IN)
  const float* w          = (const float*)d_in[1];  // (R,)
  const float* edge_val   = (const float*)d_in[2];  // (E,)
  const int*   edge_col   = (const int*)  d_in[3];  // (E,)
  // d_in[4] edge_row, d_in[5] row_node: implicit in the fixed layout
  const int*   scopes_in  = (const int*)  d_in[6];  // (N_PASS,)
  const int*   scopes_out = (const int*)  d_in[7];  // (N_PASS,)
  float* out = (float*)d_out;

  const size_t xt_bytes   = (size_t)D_IN * BATCH * sizeof(float);   // 8 MB
  const size_t logw_bytes = (size_t)R_ROWS * sizeof(float);         // 256 KB

  int use_xt = 0, have_logw = 0;
  float* xT   = nullptr;
  float* logw = nullptr;
  if (ws_size >= xt_bytes + logw_bytes) {
    use_xt = 1; have_logw = 1;
    xT   = (float*)d_ws;
    logw = (float*)((char*)d_ws + xt_bytes);
  } else if (ws_size >= xt_bytes) {
    use_xt = 1;
    xT = (float*)d_ws;
  } else if (ws_size >= logw_bytes) {
    have_logw = 1;
    logw = (float*)d_ws;
  }

  if (have_logw) {
    spn_logw<<<dim3(N_SUM / 256), 256, 0, stream>>>(w, logw);
  }
  if (use_xt) {
    spn_transpose<<<dim3(D_IN / 32, BATCH / 32), 256, 0, stream>>>(x, xT);
  }
  spn_sum_nodes<<<dim3(N_SUM / 16, BATCH / 16), 256, 0, stream>>>(
      xT, x, use_xt, w, logw, have_logw, edge_col, edge_val, out);
  spn_pass<<<dim3(N_PASS / 256, BATCH), 256, 0, stream>>>(x, scopes_in, scopes_out, out);
}